// SeasonalTrendMask_8718783611601
// MI455X (gfx1250) — compile-verified
//
#include <hip/hip_runtime.h>

// Problem constants (match reference)
#define B_      64
#define L_      2880
#define C_      128
#define PAD_    12          // K=25 -> pad 12
#define SEG_    48
#define NTSEG_  60          // L / SEG
#define TOPK_   3
#define NEG_    (-100000.0f)
#define NTILE_  (L_ / 16)   // 180
static const size_t NELEM_ = (size_t)B_ * L_ * C_;   // 23,592,960

typedef float v2f __attribute__((ext_vector_type(2)));
typedef float v4f __attribute__((ext_vector_type(4)));
typedef float v8f __attribute__((ext_vector_type(8)));

// ---------------------------------------------------------------------------
// Kernel A: NaN-row cleaning. One wave32 per (b,l) row; each lane holds 4 ch.
// ---------------------------------------------------------------------------
__global__ __launch_bounds__(256) void stm_clean(const float* __restrict__ x,
                                                 float* __restrict__ xclean) {
  int gw   = (blockIdx.x * 256 + threadIdx.x) >> 5;   // row id over B*L
  int lane = threadIdx.x & 31;
  if (gw >= B_ * L_) return;
  const float* src = x + (size_t)gw * C_ + lane * 4;
  v4f v = *(const v4f*)src;
  int isn = (v.x != v.x) | (v.y != v.y) | (v.z != v.z) | (v.w != v.w);
  if (__any(isn)) {
    v4f z = {0.f, 0.f, 0.f, 0.f};
    v = z;
  }
  *(v4f*)(xclean + (size_t)gw * C_ + lane * 4) = v;
}

// ---------------------------------------------------------------------------
// Kernel B: avgpool(K=25, count_include_pad=False) trend, xs = x - xt,
// trend mask from tmask_rand. One wave32 per (b,l) row, float4 per lane.
// x fits in L2 (94 MB < 192 MB) so the 25x read amplification is cache-hot.
// ---------------------------------------------------------------------------
__global__ __launch_bounds__(256) void stm_trend(const float* __restrict__ xclean,
                                                 const float* __restrict__ tmr,
                                                 float* __restrict__ xt_masked,
                                                 float* __restrict__ xs) {
  int gw   = (blockIdx.x * 256 + threadIdx.x) >> 5;
  int lane = threadIdx.x & 31;
  if (gw >= B_ * L_) return;
  int b = gw / L_;
  int l = gw - b * L_;
  int lo = l - PAD_; if (lo < 0) lo = 0;
  int hi = l + PAD_; if (hi > L_ - 1) hi = L_ - 1;
  float inv = 1.0f / (float)(hi - lo + 1);

  const float* rowbase = xclean + ((size_t)b * L_) * C_ + lane * 4;
  v4f s = {0.f, 0.f, 0.f, 0.f};
  for (int p = lo; p <= hi; ++p) {
    v4f v = *(const v4f*)(rowbase + (size_t)p * C_);
    s += v;
  }
  v4f xt  = s * inv;
  v4f xc  = *(const v4f*)(rowbase + (size_t)l * C_);
  v4f xsv = xc - xt;

  v4f tm = *(const v4f*)(tmr + ((size_t)(b * NTSEG_ + l / SEG_)) * C_ + lane * 4);
  v4f xtm;
  xtm.x = (tm.x < 0.2f) ? 0.0f : xt.x;
  xtm.y = (tm.y < 0.2f) ? 0.0f : xt.y;
  xtm.z = (tm.z < 0.2f) ? 0.0f : xt.z;
  xtm.w = (tm.w < 0.2f) ? 0.0f : xt.w;

  *(v4f*)(xt_masked + (size_t)gw * C_ + lane * 4) = xtm;
  *(v4f*)(xs        + (size_t)gw * C_ + lane * 4) = xsv;
}

// ---------------------------------------------------------------------------
// Zero fill (corr accumulator must start at 0 each call).
// ---------------------------------------------------------------------------
__global__ __launch_bounds__(256) void stm_zero(float* __restrict__ p, int n) {
  int i = blockIdx.x * 256 + threadIdx.x;
  if (i < n) p[i] = 0.0f;
}

// ---------------------------------------------------------------------------
// Kernel C: circular autocorr via Gram tiles with V_WMMA_F32_16X16X4_F32.
// One wave32 per 16x16 tile of G_b = X_b X_b^T (X_b : L x C).
//   A operand (16x4):  lane holds row I0+(lane&15), channels c0 + 2*(lane>>4)
//   B operand (4x16):  B[k,n] = X[J0+n, c0+k]  -> identical per-lane load.
// 32 chained WMMAs accumulate over C=128. Then 256 products are binned into
// the 31 lag diagonals via an LDS histogram, and 31 global f32 atomics.
// ---------------------------------------------------------------------------
__global__ __launch_bounds__(32) void stm_corr(const float* __restrict__ xs,
                                               float* __restrict__ corr) {
  int J0 = blockIdx.x * 16;
  int I0 = blockIdx.y * 16;
  int b  = blockIdx.z;

  int lagc = J0 - I0;
  if (lagc < 0) lagc += L_;
  // whole 31-lag span [lagc-15, lagc+15] inside the excluded edge zone?
  if (lagc <= 32 || lagc >= L_ - 32) return;

  int lane = threadIdx.x;
  int m15  = lane & 15;
  int sel  = (lane >> 4) * 2;      // K-pair select: channels {c0,c0+1} or {c0+2,c0+3}

  const float* Ap = xs + ((size_t)b * L_ + (size_t)(I0 + m15)) * C_ + sel;
  const float* Bp = xs + ((size_t)b * L_ + (size_t)(J0 + m15)) * C_ + sel;

  v8f acc = {0.f, 0.f, 0.f, 0.f, 0.f, 0.f, 0.f, 0.f};
#pragma unroll
  for (int s = 0; s < C_ / 4; ++s) {
    v2f a  = *(const v2f*)(Ap + s * 4);
    v2f bb = *(const v2f*)(Bp + s * 4);
    acc = __builtin_amdgcn_wmma_f32_16x16x4_f32(
        false, a, false, bb, (short)0, acc, false, false);
  }

  __shared__ float bins[31];
  if (lane < 31) bins[lane] = 0.0f;
  __syncthreads();

  // C/D layout: lane 0-15 -> N=lane, M=v; lane 16-31 -> N=lane-16, M=v+8
  int n     = lane & 15;
  int mbase = (lane >> 4) * 8;
#pragma unroll
  for (int v = 0; v < 8; ++v) {
    int m = mbase + v;
    atomicAdd(&bins[n - m + 15], acc[v]);   // diag index in [0,30]
  }
  __syncthreads();

  if (lane < 31) {
    int lag = lagc + lane - 15;
    if (lag < 0)    lag += L_;
    if (lag >= L_)  lag -= L_;
    atomicAdd(&corr[(size_t)b * L_ + lag], bins[lane]);
  }
}

// ---------------------------------------------------------------------------
// Kernel T: top-3 of corr per batch (with edge NEG override and /C mean),
// matching lax.top_k ordering (descending, ties -> lower index).
// ---------------------------------------------------------------------------
__global__ __launch_bounds__(256) void stm_topk(const float* __restrict__ corr,
                                                float* __restrict__ autocorr,
                                                int* __restrict__ periods) {
  int b   = blockIdx.x;
  int tid = threadIdx.x;
  __shared__ float sv[256];
  __shared__ int   si[256];
  __shared__ int   sFound[TOPK_];

  for (int k = 0; k < TOPK_; ++k) {
    float bv = -3.4e38f;
    int   bi = 0x7fffffff;
    for (int t = tid; t < L_; t += 256) {
      bool excl = false;
      for (int kk = 0; kk < k; ++kk) excl = excl || (t == sFound[kk]);
      if (excl) continue;
      float v = (t < SEG_ || t >= L_ - SEG_)
                    ? NEG_
                    : corr[(size_t)b * L_ + t] * (1.0f / (float)C_);
      if (v > bv || (v == bv && t < bi)) { bv = v; bi = t; }
    }
    sv[tid] = bv; si[tid] = bi;
    __syncthreads();
    for (int s = 128; s > 0; s >>= 1) {
      if (tid < s) {
        float ov = sv[tid + s]; int oi = si[tid + s];
        if (ov > sv[tid] || (ov == sv[tid] && oi < si[tid])) {
          sv[tid] = ov; si[tid] = oi;
        }
      }
      __syncthreads();
    }
    if (tid == 0) {
      sFound[k] = si[0];
      autocorr[b * TOPK_ + k] = sv[0];
      periods[b * TOPK_ + k]  = si[0];
    }
    __syncthreads();
  }
}

// ---------------------------------------------------------------------------
// Kernel M: season mask. One thread per (b,l,c4): read xs once, write all 3
// masked copies (k=0 is in-place on the staging region -> no alias race).
// ---------------------------------------------------------------------------
__global__ __launch_bounds__(256) void stm_season(float* __restrict__ xsbase,
                                                  const int* __restrict__ periods,
                                                  const int* __restrict__ anc) {
  int gid = blockIdx.x * 256 + threadIdx.x;
  if (gid >= B_ * L_ * (C_ / 4)) return;
  int c4  = gid & 31;
  int row = gid >> 5;
  int l   = row % L_;
  int b   = row / L_;

  size_t off = ((size_t)b * L_ + l) * C_ + (size_t)c4 * 4;
  v4f v = *(const v4f*)(xsbase + off);
  v4f z = {0.f, 0.f, 0.f, 0.f};

#pragma unroll
  for (int k = 0; k < TOPK_; ++k) {
    int p = periods[b * TOPK_ + k];     // periods.T[k][b]
    int a = anc[k * B_ + b];            // anc[k][b]
    int r = (l - a) % p;
    if (r < 0) r += p;
    bool msk = (r <= SEG_ / 2) || (r >= p - (SEG_ + 1) / 2);
    *(v4f*)(xsbase + (size_t)k * NELEM_ + off) = msk ? z : v;
  }
}

// ---------------------------------------------------------------------------
// Launch
// ---------------------------------------------------------------------------
extern "C" void kernel_launch(void* const* d_in, const int* in_sizes, int n_in,
                              void* d_out, int out_size, void* d_ws, size_t ws_size,
                              hipStream_t stream) {
  const float* x   = (const float*)d_in[0];   // [B, L, C]
  const float* tmr = (const float*)d_in[1];   // [B, NTSEG, C]
  const int*   anc = (const int*)d_in[2];     // [TOPK, B]

  float* out       = (float*)d_out;
  float* xt_masked = out;                                   // [B,L,C]
  float* autocorr  = out + NELEM_;                          // [B,TOPK]
  float* xs0       = out + NELEM_ + (size_t)B_ * TOPK_;     // xs_masked[0] (staging: xs)
  float* xclean    = xs0 + NELEM_;                          // xs_masked[1] (staging: xclean)

  float* corr    = (float*)d_ws;                            // [B, L] accumulator
  int*   periods = (int*)((float*)d_ws + (size_t)B_ * L_);  // [B, TOPK]

  int nrows = B_ * L_;
  int rowBlocks = (nrows * 32 + 255) / 256;   // 8 wave-rows per 256-thread block

  stm_clean<<<rowBlocks, 256, 0, stream>>>(x, xclean);
  stm_trend<<<rowBlocks, 256, 0, stream>>>(xclean, tmr, xt_masked, xs0);

  stm_zero<<<(B_ * L_ + 255) / 256, 256, 0, stream>>>(corr, B_ * L_);

  dim3 gcorr(NTILE_, NTILE_, B_);             // (J tile, I tile, batch) — b slowest for L2 reuse
  stm_corr<<<gcorr, 32, 0, stream>>>(xs0, corr);

  stm_topk<<<B_, 256, 0, stream>>>(corr, autocorr, periods);

  int nSeason = B_ * L_ * (C_ / 4);
  stm_season<<<(nSeason + 255) / 256, 256, 0, stream>>>(xs0, periods, anc);
}